// ScaledDotProductAttention_87557203297095
// MI455X (gfx1250) — compile-verified
//
#include <hip/hip_runtime.h>

typedef __attribute__((ext_vector_type(2))) float v2f;
typedef __attribute__((ext_vector_type(4))) float v4f;
typedef __attribute__((ext_vector_type(8))) float v8f;

#define S_DIM   2048
#define D_DIM   64
#define H_DIM   16
#define SSTRIDE 2052            // 2048 + 4 pad: conflict-free phase-3 A-frag reads, 16B aligned
#define QSCALE  0.125f
#define NEGBIG  -10000.0f

__global__ __launch_bounds__(256, 1)
void sdpa_f32_wmma_kernel(const float* __restrict__ Q, const float* __restrict__ K,
                          const float* __restrict__ V, const float* __restrict__ Mask,
                          float* __restrict__ Out, float* __restrict__ Attn)
{
    extern __shared__ float smem[];             // [32][SSTRIDE] scores/E + rowinv[32]
    float* sc     = smem;
    float* rowinv = smem + 32 * SSTRIDE;

    const int bh = blockIdx.x & 31;             // b*16 + h
    const int qt = blockIdx.x >> 5;             // 0..63  (32-row q tile)
    const int b  = bh >> 4;

    const float* q    = Q + (size_t)bh * S_DIM * D_DIM;
    const float* k    = K + (size_t)bh * S_DIM * D_DIM;
    const float* v    = V + (size_t)bh * S_DIM * D_DIM;
    const float* mask = Mask + (size_t)b * S_DIM;
    float* outp  = Out  + (size_t)bh * S_DIM * D_DIM;
    float* attnp = Attn + (size_t)bh * S_DIM * S_DIM;

    const int tid  = threadIdx.x;
    const int wave = tid >> 5;
    const int lane = tid & 31;
    const int lm   = lane & 15;                 // M (A rows) or N (B cols / C cols)
    const int kh   = (lane >> 4) * 2;           // K sub-offset within a 4-group (f32 WMMA layout)

    const int q0   = qt * 32;
    const int kmax = q0 + 32;                   // causal bound, multiple of 32
    const int nkt  = kmax >> 4;                 // number of 16-wide k tiles

    // ---------------- Phase 0: stream zeros for the beyond-causal attn region ----------------
    // softmax there underflows to exactly 0; issue these HBM stores first so STOREcnt traffic
    // overlaps the whole compute phase.
    {
        const int z4 = (S_DIM - kmax) >> 2;
        for (int idx = tid; idx < 32 * z4; idx += 256) {
            const int r = idx / z4;
            const int c = kmax + (idx - r * z4) * 4;
            v4f zz = {};
            *(v4f*)(attnp + (size_t)(q0 + r) * S_DIM + c) = zz;
        }
    }

    // ---------------- Phase 1: S = (Q*scale) K^T + masks  ->  LDS ----------------
    {
        const int   rt   = wave & 1;            // which 16-row tile of the 32-row block
        const int   qrow = q0 + rt * 16;
        const float* qp  = q + (size_t)(qrow + lm) * D_DIM + kh;

        v2f qa[16];                              // half of the 64-wide Q row, per A-frag layout
        #pragma unroll
        for (int kc = 0; kc < 16; ++kc) {
            v2f t = *(const v2f*)(qp + kc * 4);
            qa[kc].x = t.x * QSCALE;
            qa[kc].y = t.y * QSCALE;
        }

        for (int kt = (wave >> 1); kt < nkt; kt += 4) {
            const float* kp = k + (size_t)(kt * 16 + lm) * D_DIM + kh;
            v8f acc = {};
            #pragma unroll
            for (int kc = 0; kc < 16; ++kc) {
                v2f bf = *(const v2f*)(kp + kc * 4);
                acc = __builtin_amdgcn_wmma_f32_16x16x4_f32(
                          false, qa[kc], false, bf, (short)0, acc, false, false);
            }
            const int kidx = kt * 16 + lm;
            const float madd = (1.0f - mask[kidx]) * NEGBIG;
            #pragma unroll
            for (int r = 0; r < 8; ++r) {
                const int M0 = (lane < 16) ? r : (r + 8);
                float s = acc[r] + madd;
                if (kidx > qrow + M0) s += NEGBIG;     // causal
                sc[(rt * 16 + M0) * SSTRIDE + kidx] = s;
            }
        }
    }
    __syncthreads();

    // ---------------- Phase 2: fused softmax (8 lanes per row, float4 scans) ----------------
    // exp computed ONCE: E=exp(s-m) stored back to LDS for P*V; attn written as E*inv;
    // P*V normalization deferred to the phase-3 epilogue via rowinv[].
    {
        const int row = tid >> 3;               // 0..31
        const int sub = tid & 7;
        float* srow = sc + row * SSTRIDE;

        float m = -3.0e38f;
        for (int c = sub * 4; c < kmax; c += 32) {
            v4f x = *(const v4f*)(srow + c);
            m = fmaxf(fmaxf(fmaxf(m, x.x), fmaxf(x.y, x.z)), x.w);
        }
        m = fmaxf(m, __shfl_xor(m, 1, 32));
        m = fmaxf(m, __shfl_xor(m, 2, 32));
        m = fmaxf(m, __shfl_xor(m, 4, 32));

        float sum = 0.0f;
        for (int c = sub * 4; c < kmax; c += 32) {
            v4f x = *(const v4f*)(srow + c);
            v4f e;
            e.x = __expf(x.x - m);
            e.y = __expf(x.y - m);
            e.z = __expf(x.z - m);
            e.w = __expf(x.w - m);
            *(v4f*)(srow + c) = e;              // unnormalized E for P*V
            sum += (e.x + e.y) + (e.z + e.w);
        }
        sum += __shfl_xor(sum, 1, 32);
        sum += __shfl_xor(sum, 2, 32);
        sum += __shfl_xor(sum, 4, 32);
        const float inv = 1.0f / sum;

        float* arow = attnp + (size_t)(q0 + row) * S_DIM;
        for (int c = sub * 4; c < kmax; c += 32) {
            v4f e = *(const v4f*)(srow + c);
            v4f p;
            p.x = e.x * inv; p.y = e.y * inv; p.z = e.z * inv; p.w = e.w * inv;
            *(v4f*)(arow + c) = p;              // single HBM write of attn
        }
        if (sub == 0) rowinv[row] = inv;
    }
    __syncthreads();

    // ---------------- Phase 3: Out = (E * V) * rowinv  (A from LDS, B from global/L2) -------
    {
        const int rt = wave >> 2;               // 0..1 : row tile
        const int nt = wave & 3;                // 0..3 : 16-wide slice of D
        const float* vcol = v + nt * 16 + lm;
        const float* prow = sc + (rt * 16 + lm) * SSTRIDE + kh;
        v8f oacc = {};
        for (int kb = 0; kb < kmax; kb += 4) {
            v2f af = *(const v2f*)(prow + kb);
            v2f bf;
            bf.x = vcol[(size_t)(kb + kh)     * D_DIM];
            bf.y = vcol[(size_t)(kb + kh + 1) * D_DIM];
            oacc = __builtin_amdgcn_wmma_f32_16x16x4_f32(
                       false, af, false, bf, (short)0, oacc, false, false);
        }
        #pragma unroll
        for (int r = 0; r < 8; ++r) {
            const int M0 = (lane < 16) ? r : (r + 8);
            const float invr = rowinv[rt * 16 + M0];
            outp[(size_t)(q0 + rt * 16 + M0) * D_DIM + nt * 16 + lm] = oacc[r] * invr;
        }
    }
}

extern "C" void kernel_launch(void* const* d_in, const int* in_sizes, int n_in,
                              void* d_out, int out_size, void* d_ws, size_t ws_size,
                              hipStream_t stream) {
    const float* Q = (const float*)d_in[0];
    const float* K = (const float*)d_in[1];
    const float* V = (const float*)d_in[2];
    const float* M = (const float*)d_in[3];
    float* Out  = (float*)d_out;
    float* Attn = Out + (size_t)2 * H_DIM * S_DIM * D_DIM;   // tuple: (output, attn)

    const size_t shmem = (size_t)(32 * SSTRIDE + 32) * sizeof(float);  // ~262.8 KB of 320 KB LDS
    dim3 grid(2 * H_DIM * (S_DIM / 32));   // 2048 workgroups: (b,h) x 64 q-tiles
    dim3 block(256);                        // 8 waves (wave32)
    sdpa_f32_wmma_kernel<<<grid, block, shmem, stream>>>(Q, K, V, M, Out, Attn);
}